// EquivariantUpsampleConv_51926154609209
// MI455X (gfx1250) — compile-verified
//
#include <hip/hip_runtime.h>
#include <hip/hip_bf16.h>

typedef __attribute__((ext_vector_type(2))) float v2f;
typedef __attribute__((ext_vector_type(8))) float v8f;

#define HR     512
#define NPIX   (HR * HR)        // 262144
#define CTOT   22
#define KSTEPS 121              // 484 / 4
#define WPB    8                // waves per block
#define TILES  (NPIX / 16)      // 16384
#define NBLK   (TILES / WPB)    // 2048

__device__ __forceinline__ double fctl(int n) {
    double r = 1.0;
    for (int i = 2; i <= n; ++i) r *= (double)i;
    return r;
}

struct C2 { double re, im; };

// Row i of the real->complex change-of-basis Q_l, including the (-i)^l prefactor
// ((-i)^4 = +1, (-i)^6 = -1, both real for even l). Returns #nonzeros (1 or 2).
__device__ __forceinline__ int qrow(int l, int i, int* cols, C2* vals) {
    const int m = i - l;
    const double s = (l == 6) ? -1.0 : 1.0;
    const double r = 0.70710678118654752440;
    if (m < 0) {
        cols[0] = l - m; vals[0].re = s * r; vals[0].im = 0.0;
        cols[1] = l + m; vals[1].re = 0.0;   vals[1].im = -s * r;
        return 2;
    } else if (m == 0) {
        cols[0] = l; vals[0].re = s; vals[0].im = 0.0;
        return 1;
    } else {
        const double sg = (m & 1) ? -1.0 : 1.0;
        cols[0] = l + m; vals[0].re = s * sg * r; vals[0].im = 0.0;
        cols[1] = l - m; vals[1].re = 0.0;        vals[1].im = s * sg * r;
        return 2;
    }
}

__global__ __launch_bounds__(256)
void equi_upconv_tp_kernel(const float* __restrict__ f4,
                           const float* __restrict__ f6,
                           const float* __restrict__ sw,
                           const float* __restrict__ tpw,
                           float* __restrict__ dout)
{
    __shared__ float Wlds[CTOT * CTOT * CTOT];   // Wbig[i][j][k], 42592 B

    const int tid = threadIdx.x;

    // ---- stage 0: clear LDS tensor ----
    for (int i = tid; i < CTOT * CTOT * CTOT; i += 256) Wlds[i] = 0.0f;
    __syncthreads();

    // ---- stage 1: build combined bilinear tensor from CG coefficients ----
    // PATHS = (4,4,4)(4,4,6)(4,6,4)(4,6,6)(6,4,4)(6,4,6)(6,6,4)(6,6,6)
    {
        const int LA[8] = {4,4,4,4,6,6,6,6};
        const int LB[8] = {4,4,6,6,4,4,6,6};
        const int LC[8] = {4,6,4,6,4,6,4,6};
        const int off[9] = {0,81,162,279,396,513,630,799,968};

        for (int item = tid; item < 968; item += 256) {
            int p = 0;
            while (item >= off[p + 1]) ++p;
            const int la = LA[p], lb = LB[p], lc = LC[p];
            const int rem = item - off[p];
            const int nb = 2 * lb + 1;
            const int ia = rem / nb, ib = rem % nb;
            const int m1 = ia - la, m2 = ib - lb, m3 = m1 + m2;
            if (m3 < -lc || m3 > lc) continue;

            // Clebsch-Gordan coefficient
            double pre = sqrt((double)(2 * lc + 1) *
                              fctl(lc + la - lb) * fctl(lc - la + lb) * fctl(la + lb - lc) /
                              fctl(la + lb + lc + 1));
            pre *= sqrt(fctl(lc + m3) * fctl(lc - m3) * fctl(la - m1) *
                        fctl(la + m1) * fctl(lb - m2) * fctl(lb + m2));
            double ssum = 0.0;
            for (int k = 0; k <= la + lb - lc; ++k) {
                const int d1 = la + lb - lc - k, d2 = la - m1 - k, d3 = lb + m2 - k;
                const int d4 = lc - lb + m1 + k, d5 = lc - la - m2 + k;
                if (d1 < 0 || d2 < 0 || d3 < 0 || d4 < 0 || d5 < 0) continue;
                const double den = fctl(k) * fctl(d1) * fctl(d2) * fctl(d3) * fctl(d4) * fctl(d5);
                ssum += ((k & 1) ? -1.0 : 1.0) / den;
            }
            const double cg = pre * ssum;
            if (cg == 0.0) continue;

            // alpha*W3J = (sqrt(2lc+1)/2) * (Cr/sqrt(2lc+1)) = 0.5*Cr
            const double scale = 0.5 * (double)tpw[p] * cg;
            const int ic = lc + m3;

            int c1[2]; C2 v1[2]; const int n1 = qrow(la, ia, c1, v1);
            int c2[2]; C2 v2[2]; const int n2 = qrow(lb, ib, c2, v2);
            int c3[2]; C2 v3[2]; const int n3 = qrow(lc, ic, c3, v3);
            const int oa = (la == 6) ? 9 : 0;
            const int ob = (lb == 6) ? 9 : 0;
            const int oc = (lc == 6) ? 9 : 0;

            for (int a = 0; a < n1; ++a)
                for (int b = 0; b < n2; ++b)
                    for (int c = 0; c < n3; ++c) {
                        const double re12 = v1[a].re * v2[b].re - v1[a].im * v2[b].im;
                        const double im12 = v1[a].re * v2[b].im + v1[a].im * v2[b].re;
                        const double re = re12 * v3[c].re + im12 * v3[c].im;  // * conj(q3)
                        const float add = (float)(re * scale);
                        if (add != 0.0f) {
                            const int idx = ((oa + c1[a]) * CTOT + (ob + c2[b])) * CTOT + (oc + c3[c]);
                            atomicAdd(&Wlds[idx], add);   // ds_add_f32
                        }
                    }
        }
    }
    __syncthreads();

    // ---- stage 2: per-pixel x / y, then WMMA GEMM (N x 484) x (484 x 22) ----
    const int lane = tid & 31;
    const int wave = tid >> 5;
    const bool hi  = lane >= 16;
    const int mcol = lane & 15;
    const int tile = blockIdx.x * WPB + wave;
    const int npix = tile * 16 + mcol;
    const int Hh = npix >> 9, Ww = npix & 511;

    float k3[9];
#pragma unroll
    for (int i = 0; i < 9; ++i) k3[i] = sw[i];

    float x[22], y[22];
    {
        const int lr = (Hh >> 2) * 128 + (Ww >> 2);
#pragma unroll
        for (int c = 0; c < 9; ++c)  x[c]     = f4[lr * 9 + c];
#pragma unroll
        for (int c = 0; c < 13; ++c) x[9 + c] = f6[lr * 13 + c];
    }
    {
        // 3x3 edge-padded window on the 4x-upsampled grid hits at most 2x2
        // distinct low-res sources; fold kernel taps into per-source weights.
        const int hm = (Hh > 0   ? Hh - 1 : 0)   >> 2;
        const int hp = (Hh < 511 ? Hh + 1 : 511) >> 2;
        const int h0 = Hh >> 2;
        const int wm = (Ww > 0   ? Ww - 1 : 0)   >> 2;
        const int wp = (Ww < 511 ? Ww + 1 : 511) >> 2;
        const int w0 = Ww >> 2;
        const int selH[3] = {0, (h0 == hm) ? 0 : 1, (hp == hm) ? 0 : 1};
        const int selW[3] = {0, (w0 == wm) ? 0 : 1, (wp == wm) ? 0 : 1};
        float W2[4] = {0.f, 0.f, 0.f, 0.f};
#pragma unroll
        for (int dh = 0; dh < 3; ++dh)
#pragma unroll
            for (int dw = 0; dw < 3; ++dw)
                W2[selH[dh] * 2 + selW[dw]] += k3[dh * 3 + dw];
        const int rows[2] = {hm, hp}, colsrc[2] = {wm, wp};
#pragma unroll
        for (int c = 0; c < 22; ++c) y[c] = 0.0f;
#pragma unroll
        for (int s = 0; s < 2; ++s)
#pragma unroll
            for (int t = 0; t < 2; ++t) {
                const float w = W2[s * 2 + t];
                const int src = rows[s] * 128 + colsrc[t];
#pragma unroll
                for (int c = 0; c < 9; ++c)  y[c]     += w * f4[src * 9 + c];
#pragma unroll
                for (int c = 0; c < 13; ++c) y[9 + c] += w * f6[src * 13 + c];
            }
    }

    // A (16x4 f32): lanes 0-15 carry K=k0,k0+1; lanes 16-31 carry K=k0+2,k0+3.
    // B (4x16 f32): lane n<16 col n for K=kb,kb+1 (vgpr0/1); lanes 16-31 K=kb+2,kb+3.
    v8f acc0 = {0.f, 0.f, 0.f, 0.f, 0.f, 0.f, 0.f, 0.f};
    v8f acc1 = {0.f, 0.f, 0.f, 0.f, 0.f, 0.f, 0.f, 0.f};
    const int n0 = mcol;
    const int n1c = (16 + mcol < 22) ? (16 + mcol) : 21;   // clamp; cols>=22 never stored

#pragma unroll
    for (int t = 0; t < KSTEPS; ++t) {
        const int k0 = 4 * t;
        const float p0 = x[(k0    ) / 22] * y[(k0    ) % 22];
        const float p1 = x[(k0 + 1) / 22] * y[(k0 + 1) % 22];
        const float p2 = x[(k0 + 2) / 22] * y[(k0 + 2) % 22];
        const float p3 = x[(k0 + 3) / 22] * y[(k0 + 3) % 22];
        v2f a;
        a.x = hi ? p2 : p0;
        a.y = hi ? p3 : p1;
        const int kb = k0 + (hi ? 2 : 0);
        v2f b0, b1;
        b0.x = Wlds[(kb    ) * CTOT + n0];
        b0.y = Wlds[(kb + 1) * CTOT + n0];
        b1.x = Wlds[(kb    ) * CTOT + n1c];
        b1.y = Wlds[(kb + 1) * CTOT + n1c];
        acc0 = __builtin_amdgcn_wmma_f32_16x16x4_f32(false, a, false, b0, (short)0, acc0, false, false);
        acc1 = __builtin_amdgcn_wmma_f32_16x16x4_f32(false, a, false, b1, (short)0, acc1, false, false);
    }

    // ---- stage 3: residual + store. D layout: lane -> col, vgpr r -> row r+8*hi.
#pragma unroll
    for (int r = 0; r < 8; ++r) {
        const int prow = r + (hi ? 8 : 0);
        const int np = tile * 16 + prow;
        const int lr = ((np >> 9) >> 2) * 128 + (((np & 511)) >> 2);
        {
            const int k = mcol;   // 0..15
            const float xv = (k < 9) ? f4[lr * 9 + k] : f6[lr * 13 + (k - 9)];
            const float v = acc0[r] + xv;
            if (k < 9) dout[np * 9 + k] = v;
            else       dout[NPIX * 9 + np * 13 + (k - 9)] = v;
        }
        const int k1 = 16 + mcol;
        if (k1 < 22) {
            const float xv = f6[lr * 13 + (k1 - 9)];
            dout[NPIX * 9 + np * 13 + (k1 - 9)] = acc1[r] + xv;
        }
    }
}

extern "C" void kernel_launch(void* const* d_in, const int* in_sizes, int n_in,
                              void* d_out, int out_size, void* d_ws, size_t ws_size,
                              hipStream_t stream) {
    (void)in_sizes; (void)n_in; (void)out_size; (void)d_ws; (void)ws_size;
    const float* f4  = (const float*)d_in[0];
    const float* f6  = (const float*)d_in[1];
    const float* sw  = (const float*)d_in[2];
    const float* tpw = (const float*)d_in[3];
    float* out = (float*)d_out;
    equi_upconv_tp_kernel<<<NBLK, 256, 0, stream>>>(f4, f6, sw, tpw, out);
}